// VertexGNNEncoder_39702677684854
// MI455X (gfx1250) — compile-verified
//
#include <hip/hip_runtime.h>
#include <hip/hip_bf16.h>

// ---------------------------------------------------------------------------
// GraphSAGE (mean) x2 + linear head for MI455X (gfx1250, wave32).
//
//  h1 = relu( mean_aggr(x)  @ W1_l + b1 + x  @ W1_r )
//  h2 = relu( mean_aggr(h1) @ W2_l + b2 + h1 @ W2_r )
//  out = h2 @ W_lin + b_lin
//
// Aggregation = gather + fp32 global atomics (102 MB working set is L2
// resident on the 192 MB L2). Dense parts use V_WMMA_F32_16X16X4_F32 (full
// fp32 precision, bit-compatible with the fp32 reference), with A tiles
// staged to LDS via GLOBAL_LOAD_ASYNC_TO_LDS_B128 (ASYNCcnt) and B fragments
// reused across 5 M-tiles per wave.
// ---------------------------------------------------------------------------

typedef __attribute__((ext_vector_type(2))) float v2f;
typedef __attribute__((ext_vector_type(8))) float v8f;
typedef int v4i __attribute__((vector_size(16)));   // matches builtin proto

#define FEAT 128            // feature width (K dim of every GEMM)
#define MT   5              // M tiles per block: 80 rows; 100000 = 80 * 1250
#define ROWS (MT * 16)      // 80
#define LSTR 132            // padded LDS row stride (floats): 132 % 64 = 4
                            // -> 16 row-reads at one column hit 16 banks

// --- CDNA5 async global->LDS staging (ASYNCcnt) with safe fallback ---------
#if defined(__AMDGCN__) && __has_builtin(__builtin_amdgcn_global_load_async_to_lds_b128)
#define USE_ASYNC_STAGE 1
#else
#define USE_ASYNC_STAGE 0
#endif

#if USE_ASYNC_STAGE
typedef __attribute__((address_space(1))) v4i* gptr_b128;  // __device__ int4*
typedef __attribute__((address_space(3))) v4i* lptr_b128;  // __shared__ int4*
#endif

__device__ __forceinline__ void stage16(float* lds_dst, const float* gsrc) {
#if USE_ASYNC_STAGE
  __builtin_amdgcn_global_load_async_to_lds_b128(
      (gptr_b128)(size_t)gsrc, (lptr_b128)(size_t)(lds_dst), 0, 0);
#else
  *(float4*)lds_dst = *(const float4*)gsrc;
#endif
}

__device__ __forceinline__ void stage_join() {
#if USE_ASYNC_STAGE
#if __has_builtin(__builtin_amdgcn_s_wait_asynccnt)
  __builtin_amdgcn_s_wait_asynccnt(0);
#else
  asm volatile("s_wait_asynccnt 0" ::: "memory");
#endif
#endif
}

__device__ __forceinline__ v8f wmma4(v2f a, v2f b, v8f c) {
  return __builtin_amdgcn_wmma_f32_16x16x4_f32(false, a, false, b,
                                               (short)0, c, false, false);
}

__device__ __forceinline__ void atomAddF32(float* p, float v) {
  unsafeAtomicAdd(p, v);   // lowers to global_atomic_add_f32 (no CAS loop)
}

// ---------------------------------------------------------------------------
// Degree: one thread per edge, deg[dst] += 1.0f; then reciprocal in place.
// ---------------------------------------------------------------------------
__global__ void degree_kernel(const long long* __restrict__ dst,
                              float* __restrict__ deg, int E) {
  int e = blockIdx.x * blockDim.x + threadIdx.x;
  if (e < E) atomAddF32(deg + dst[e], 1.0f);
}

__global__ void invdeg_kernel(float* __restrict__ deg, int N) {
  int i = blockIdx.x * blockDim.x + threadIdx.x;
  if (i < N) deg[i] = 1.0f / fmaxf(deg[i], 1.0f);
}

// ---------------------------------------------------------------------------
// Scatter-add: one wave per edge. Each of the 32 lanes moves one float4
// (32 * 16B = 512B = one feature row) and issues 4 fp32 atomics. Random dst
// rows land in the L2-resident accumulator.
// ---------------------------------------------------------------------------
__global__ __launch_bounds__(256)
void scatter_kernel(const float* __restrict__ feat,
                    const long long* __restrict__ src,
                    const long long* __restrict__ dst,
                    float* __restrict__ acc, int E) {
  const int e = blockIdx.x * 8 + (threadIdx.x >> 5);
  if (e >= E) return;
  const int lane = threadIdx.x & 31;
  const long long s = src[e];
  const long long d = dst[e];
  const float4 v = ((const float4*)(feat + (size_t)s * FEAT))[lane];
  float* p = acc + (size_t)d * FEAT + lane * 4;
  atomAddF32(p + 0, v.x);
  atomAddF32(p + 1, v.y);
  atomAddF32(p + 2, v.z);
  atomAddF32(p + 3, v.w);
}

// ---------------------------------------------------------------------------
// Fused SAGE layer via fp32 WMMA:
//   out[80 x Nout] = act( (acc*inv_deg) @ Wl + bias + xin @ Wr )
//
// Block = (Nout/16) waves; wave w owns output columns [16w, 16w+16).
// Block covers 80 rows = 5 M-tiles. x/acc tiles staged to LDS (async),
// scale-by-inv_deg applied at fragment load. Per K4 step a wave loads the
// B fragments once and feeds 5 (HAS_AGGR: 10) WMMAs -> 5 independent
// accumulator chains for the scheduler.
//
// Fragment layouts (ISA 7.12.2, wave32):
//   A (16x4 f32):  lane%16 = M row; lane/16 selects K pair {0,1}/{2,3}.
//   B (4x16 f32):  lane%16 = N col; same K-pair split across lane halves.
//   C/D (16x16):   VGPR r: lanes 0-15 -> M=r, lanes 16-31 -> M=r+8; N=lane%16.
// ---------------------------------------------------------------------------
template <int HAS_AGGR, int RELU>
__global__ __launch_bounds__(256)
void sage_layer_kernel(const float* __restrict__ xin,     // [N, FEAT]
                       const float* __restrict__ acc,     // [N, FEAT] (HAS_AGGR)
                       const float* __restrict__ inv_deg, // [N]       (HAS_AGGR)
                       const float* __restrict__ Wl,      // [FEAT, Nout] (HAS_AGGR)
                       const float* __restrict__ bias,    // [Nout]
                       const float* __restrict__ Wr,      // [FEAT, Nout]
                       float* __restrict__ out,           // [N, Nout]
                       int Nout) {
  __shared__ alignas(16) float lds_x[ROWS * LSTR];
  __shared__ alignas(16) float lds_a[HAS_AGGR ? ROWS * LSTR : 4];

  const int wave   = threadIdx.x >> 5;
  const int lane   = threadIdx.x & 31;
  const int nwaves = blockDim.x >> 5;
  const int row0   = blockIdx.x * ROWS;

  // ---- stage A tiles (x, and acc when aggregating) into LDS -------------
  // copy i<ROWS: x row i; copy i>=ROWS: acc row i-ROWS. One wave per row:
  // 32 lanes x 16B = one 512B feature row. i is wave-uniform.
  const int ncopies = HAS_AGGR ? 2 * ROWS : ROWS;
  for (int i = wave; i < ncopies; i += nwaves) {
    if (HAS_AGGR && i >= ROWS) {
      const int r = i - ROWS;
      stage16(lds_a + r * LSTR + lane * 4,
              acc + (size_t)(row0 + r) * FEAT + lane * 4);
    } else {
      stage16(lds_x + i * LSTR + lane * 4,
              xin + (size_t)(row0 + i) * FEAT + lane * 4);
    }
  }
  stage_join();
  __syncthreads();

  const int m  = lane & 15;   // A: M row | B,C: N col
  const int kh = lane >> 4;   // K pair within each K4 step
  const int n0 = wave * 16;

  float sc[MT];
  if (HAS_AGGR) {
#pragma unroll
    for (int t = 0; t < MT; ++t) sc[t] = inv_deg[row0 + t * 16 + m];
  }

  // C init: every C register holds column n0+m -> broadcast bias.
  const float bz = bias[n0 + m];
  v8f c[MT];
#pragma unroll
  for (int t = 0; t < MT; ++t)
#pragma unroll
    for (int r = 0; r < 8; ++r) c[t][r] = bz;

  for (int k0 = 0; k0 < FEAT; k0 += 4) {
    const int ka = k0 + 2 * kh;

    v2f br;   // B fragment of Wr, reused by MT WMMAs
    br.x = Wr[(size_t)ka * Nout + n0 + m];
    br.y = Wr[(size_t)(ka + 1) * Nout + n0 + m];
    v2f bl;
    if (HAS_AGGR) {
      bl.x = Wl[(size_t)ka * Nout + n0 + m];
      bl.y = Wl[(size_t)(ka + 1) * Nout + n0 + m];
    }

#pragma unroll
    for (int t = 0; t < MT; ++t) {
      const int rb = t * 16 + m;
      v2f ax = *(const v2f*)&lds_x[rb * LSTR + ka];       // ds_load_b64
      c[t] = wmma4(ax, br, c[t]);
      if (HAS_AGGR) {
        v2f aa = *(const v2f*)&lds_a[rb * LSTR + ka];
        aa.x *= sc[t];
        aa.y *= sc[t];
        c[t] = wmma4(aa, bl, c[t]);
      }
    }
  }

#pragma unroll
  for (int t = 0; t < MT; ++t)
#pragma unroll
    for (int r = 0; r < 8; ++r) {
      float v = c[t][r];
      if (RELU) v = fmaxf(v, 0.0f);
      out[(size_t)(row0 + t * 16 + r + 8 * kh) * Nout + n0 + m] = v;
    }
}

// ---------------------------------------------------------------------------
// Host-side orchestration (graph-capture safe: only async ops on `stream`).
// ---------------------------------------------------------------------------
extern "C" void kernel_launch(void* const* d_in, const int* in_sizes, int n_in,
                              void* d_out, int out_size, void* d_ws, size_t ws_size,
                              hipStream_t stream) {
  const float*     x    = (const float*)d_in[0];
  const long long* ei   = (const long long*)d_in[1];  // int64 [2, E]
  const float*     W1l  = (const float*)d_in[2];
  const float*     b1   = (const float*)d_in[3];
  const float*     W1r  = (const float*)d_in[4];
  const float*     W2l  = (const float*)d_in[5];
  const float*     b2   = (const float*)d_in[6];
  const float*     W2r  = (const float*)d_in[7];
  const float*     Wlin = (const float*)d_in[8];
  const float*     blin = (const float*)d_in[9];
  float*           out  = (float*)d_out;

  const int N = in_sizes[0] / FEAT;   // 100000
  const int E = in_sizes[1] / 2;      // 1600000
  const long long* src = ei;          // row 0: message sources
  const long long* dst = ei + E;      // row 1: aggregation targets

  // Workspace layout: inv_deg | acc | h1 | h2  (~154 MB)
  float* inv  = (float*)d_ws;
  float* accb = inv + N;                      // 16B-aligned (N*4 % 16 == 0)
  float* h1   = accb + (size_t)N * FEAT;
  float* h2   = h1 + (size_t)N * FEAT;

  const int edge_blocks = (E + 255) / 256;
  const int wavescatter = (E + 7) / 8;        // 8 waves (edges) per block
  const int node_blocks = (N + 255) / 256;
  const int tile_blocks = N / ROWS;           // 1250 (exact)

  // --- degrees -> reciprocal (shared by both layers) ---
  (void)hipMemsetAsync(inv, 0, (size_t)N * sizeof(float), stream);
  degree_kernel<<<edge_blocks, 256, 0, stream>>>(dst, inv, E);
  invdeg_kernel<<<node_blocks, 256, 0, stream>>>(inv, N);

  // --- layer 1 ---
  (void)hipMemsetAsync(accb, 0, (size_t)N * FEAT * sizeof(float), stream);
  scatter_kernel<<<wavescatter, 256, 0, stream>>>(x, src, dst, accb, E);
  sage_layer_kernel<1, 1><<<tile_blocks, 256, 0, stream>>>(
      x, accb, inv, W1l, b1, W1r, h1, FEAT);

  // --- layer 2 ---
  (void)hipMemsetAsync(accb, 0, (size_t)N * FEAT * sizeof(float), stream);
  scatter_kernel<<<wavescatter, 256, 0, stream>>>(h1, src, dst, accb, E);
  sage_layer_kernel<1, 1><<<tile_blocks, 256, 0, stream>>>(
      h1, accb, inv, W2l, b2, W2r, h2, FEAT);

  // --- linear head: h2 @ W_lin + b_lin (Nout=64 -> 4 waves per block) ---
  sage_layer_kernel<0, 0><<<tile_blocks, 128, 0, stream>>>(
      h2, nullptr, nullptr, nullptr, blin, Wlin, out, 64);
}